// QuantumLayerVQC_45749991637439
// MI455X (gfx1250) — compile-verified
//
#include <hip/hip_runtime.h>
#include <hip/hip_bf16.h>
#include <math.h>

typedef __attribute__((ext_vector_type(2))) float v2f;
typedef __attribute__((ext_vector_type(8))) float v8f;

#define NQ 3
#define NL 3

// ---------------------------------------------------------------------------
// Setup: build the fixed 8x8 real-symmetric effective observable
//   M = Re( U^dagger Z0 U ),  U = variational circuit from q_weights.
// Threads 0..7 each propagate one computational-basis column through the
// exact complex circuit; threads 0..63 then contract M_ij. Output: 64 floats.
// ---------------------------------------------------------------------------
__global__ void vqc_build_M(const float* __restrict__ w, float* __restrict__ M) {
  __shared__ float Ure[8][8];
  __shared__ float Uim[8][8];
  const int t = threadIdx.x;
  if (t < 8) {
    float re[8], im[8];
#pragma unroll
    for (int k = 0; k < 8; ++k) { re[k] = (k == t) ? 1.f : 0.f; im[k] = 0.f; }
#pragma unroll
    for (int l = 0; l < NL; ++l) {
#pragma unroll
      for (int q = 0; q < NQ; ++q) {
        const int bit = 4 >> q;
        // RY(q, w[l][q][0])
        float c, s;
        sincosf(0.5f * w[(l * NQ + q) * 2 + 0], &s, &c);
#pragma unroll
        for (int k = 0; k < 8; ++k) {
          if (!(k & bit)) {
            const int k1 = k | bit;
            float r0 = re[k], i0 = im[k], r1 = re[k1], i1 = im[k1];
            re[k]  = c * r0 - s * r1;  im[k]  = c * i0 - s * i1;
            re[k1] = s * r0 + c * r1;  im[k1] = s * i0 + c * i1;
          }
        }
        // RZ(q, w[l][q][1]):  bit==0 -> *exp(-i phi/2), bit==1 -> *exp(+i phi/2)
        float cp, sp;
        sincosf(0.5f * w[(l * NQ + q) * 2 + 1], &sp, &cp);
#pragma unroll
        for (int k = 0; k < 8; ++k) {
          const float r = re[k], i = im[k];
          if (k & bit) { re[k] = r * cp - i * sp; im[k] = i * cp + r * sp; }
          else         { re[k] = r * cp + i * sp; im[k] = i * cp - r * sp; }
        }
      }
      // CNOT(0,1) then CNOT(1,2)
#pragma unroll
      for (int cq = 0; cq < NQ - 1; ++cq) {
        const int cb = 4 >> cq, tb = 4 >> (cq + 1);
#pragma unroll
        for (int k = 0; k < 8; ++k) {
          if ((k & cb) && !(k & tb)) {
            const int k1 = k | tb;
            float tr = re[k]; re[k] = re[k1]; re[k1] = tr;
            float ti = im[k]; im[k] = im[k1]; im[k1] = ti;
          }
        }
      }
    }
#pragma unroll
    for (int k = 0; k < 8; ++k) { Ure[k][t] = re[k]; Uim[k][t] = im[k]; }
  }
  __syncthreads();
  // M_ij = sum_k z_k * Re(conj(U_ki) U_kj),  z_k = +1 if qubit0(k)==0 else -1
  const int i = t >> 3, j = t & 7;
  float acc = 0.f;
#pragma unroll
  for (int k = 0; k < 8; ++k) {
    const float z = (k & 4) ? -1.f : 1.f;
    acc += z * (Ure[k][i] * Ure[k][j] + Uim[k][i] * Uim[k][j]);
  }
  M[t] = acc;
}

// ---------------------------------------------------------------------------
// WMMA helper: D = A(16x4,f32) * B(4x16,f32) + C   (V_WMMA_F32_16X16X4_F32)
// ---------------------------------------------------------------------------
__device__ __forceinline__ v8f wmma_f32_4(v2f a, v2f b, v8f c) {
  return __builtin_amdgcn_wmma_f32_16x16x4_f32(
      /*neg_a=*/false, a, /*neg_b=*/false, b,
      /*c_mod=*/(short)0, c, /*reuse_a=*/false, /*reuse_b=*/false);
}

// ---------------------------------------------------------------------------
// Main: each wave handles 32 batch rows.
//   s[abc] = f_a(x0) f_b(x1) f_c(x2)  (real product state after encoding)
//   T = S(32x8) * Mext(8x16)  via 4x v_wmma_f32_16x16x4_f32
//   out = sum_n s[n]*T[n]
// All WMMA issued with full EXEC (tail handled by index clamp; only the final
// store is predicated).
// ---------------------------------------------------------------------------
__global__ __launch_bounds__(256) void vqc_eval(const float* __restrict__ x,
                                                const float* __restrict__ Mg,
                                                float* __restrict__ out,
                                                int batch) {
  __shared__ float sM[8][16];    // M padded to 8x16 (cols 8..15 zero)
  __shared__ float sS[8][288];   // per-wave S: [row*9 + k], stride 9 vs banks
  __shared__ float sT[8][544];   // per-wave T: [row*17 + n]

  const int tid = threadIdx.x;
  if (tid < 128) {
    const int k = tid >> 4, n = tid & 15;
    sM[k][n] = (n < 8) ? Mg[k * 8 + n] : 0.f;
  }
  __syncthreads();

  const int wave = tid >> 5, lane = tid & 31;
  const int half = lane >> 4, r = lane & 15;   // half: K/M-group select, r: N / M row
  float* __restrict__ S = sS[wave];
  float* __restrict__ T = sT[wave];

  // Loop-invariant B fragments (ISA 7.12.2: B row K -> VGPR=K&1, lane-half=K>>1)
  v2f b0, b1;
  b0.x = sM[2 * half + 0][r];  b0.y = sM[2 * half + 1][r];   // K = 0..3
  b1.x = sM[2 * half + 4][r];  b1.y = sM[2 * half + 5][r];   // K = 4..7

  const long gidx = (long)blockIdx.x * 256 + tid;
  int row = (int)gidx;
  if (row >= batch) row = batch - 1;           // clamp: keep EXEC all-ones

  // x row: [B,8] f32; only x0,x1,x2 used -> one coalesced b128 load
  const float4 xv = ((const float4*)x)[(long)row * 2];

  float c0, s0, c1, s1, c2, s2;
  sincosf(0.5f * xv.x, &s0, &c0);
  sincosf(0.5f * xv.y, &s1, &c1);
  sincosf(0.5f * xv.z, &s2, &c2);
  const float f0[2] = {c0, s0}, f1[2] = {c1, s1}, f2[2] = {c2, s2};

  float sv[8];
#pragma unroll
  for (int a = 0; a < 2; ++a)
#pragma unroll
    for (int b = 0; b < 2; ++b)
#pragma unroll
      for (int c = 0; c < 2; ++c)
        sv[a * 4 + b * 2 + c] = f0[a] * f1[b] * f2[c];

  // Stage S (wave-local; DS ops are in-order within a wave)
#pragma unroll
  for (int k = 0; k < 8; ++k) S[lane * 9 + k] = sv[k];

  // A fragments (ISA 7.12.2: lanes 0-15 = rows, VGPR pair = K{0,1}; lanes 16-31 = K{2,3})
  const int kb = 2 * half;
  v2f a00, a01, a10, a11;
  a00.x = S[r * 9 + kb];            a00.y = S[r * 9 + kb + 1];
  a01.x = S[r * 9 + 4 + kb];        a01.y = S[r * 9 + 4 + kb + 1];
  a10.x = S[(16 + r) * 9 + kb];     a10.y = S[(16 + r) * 9 + kb + 1];
  a11.x = S[(16 + r) * 9 + 4 + kb]; a11.y = S[(16 + r) * 9 + 4 + kb + 1];

  v8f acc0 = {}, acc1 = {};
  acc0 = wmma_f32_4(a00, b0, acc0);   // rows 0-15,  K 0-3
  acc0 = wmma_f32_4(a01, b1, acc0);   // rows 0-15,  K 4-7
  acc1 = wmma_f32_4(a10, b0, acc1);   // rows 16-31, K 0-3
  acc1 = wmma_f32_4(a11, b1, acc1);   // rows 16-31, K 4-7

  // Scatter D (C/D layout: VGPR v -> row v + 8*half, col = r) into T
#pragma unroll
  for (int v = 0; v < 8; ++v) {
    T[(v + 8 * half) * 17 + r]        = acc0[v];
    T[(16 + v + 8 * half) * 17 + r]   = acc1[v];
  }

  // out = s . T[row]
  float res = 0.f;
#pragma unroll
  for (int n = 0; n < 8; ++n) res += sv[n] * T[lane * 17 + n];

  if (gidx < batch) out[gidx] = res;
}

extern "C" void kernel_launch(void* const* d_in, const int* in_sizes, int n_in,
                              void* d_out, int out_size, void* d_ws, size_t ws_size,
                              hipStream_t stream) {
  const float* x = (const float*)d_in[0];        // [B, 8] f32
  const float* w = (const float*)d_in[1];        // [3, 3, 2] f32
  float* out = (float*)d_out;                    // [B, 1] f32
  float* M = (float*)d_ws;                       // 64 floats scratch

  const int batch = in_sizes[0] / 8;

  vqc_build_M<<<1, 64, 0, stream>>>(w, M);
  const int blocks = (batch + 255) / 256;
  vqc_eval<<<blocks, 256, 0, stream>>>(x, M, out, batch);
}